// AttentionWithContext_15212774163146
// MI455X (gfx1250) — compile-verified
//
#include <hip/hip_runtime.h>

typedef float v2f __attribute__((ext_vector_type(2)));
typedef float v8f __attribute__((ext_vector_type(8)));

// D = A(16x4) x B(4x16) + C(16x16), full fp32 precision on the matrix pipe.
__device__ __forceinline__ v8f wmma4(v2f a, v2f b, v8f c) {
    return __builtin_amdgcn_wmma_f32_16x16x4_f32(
        false, a, false, b, (short)0, c, false, false);
}

// ---------------------------------------------------------------------------
// Generic fp32 GEMM:  C[M,N] = A[M,K] @ W[N,K]^T  (both row-major)
// Block tile 128x64, K-step 16, 256 threads = 8 waves in a 4x2 grid,
// each wave computes a 32x32 tile as 2x2 WMMA 16x16 tiles.
// MODE 0: plain row-major output (+ optional bias)
// MODE 1: fused QKV scatter into Q(B,H,N,D), K/V(B,H,256+N,D)
// ---------------------------------------------------------------------------
template <int MODE>
__global__ __launch_bounds__(256) void gemm_kernel(
    const float* __restrict__ A, const float* __restrict__ W,
    const float* __restrict__ bias, float* __restrict__ out,
    float* __restrict__ Qb, float* __restrict__ Kb, float* __restrict__ Vb,
    int Ncols, int Kdim)
{
    __shared__ float As[128 * 20];   // stride 20 floats: 16B-aligned float4, conflict-free b64 reads
    __shared__ float Wsh[64 * 20];

    const int tid  = threadIdx.x;
    const int wid  = tid >> 5;
    const int lane = tid & 31;
    const int half = lane >> 4;
    const int l15  = lane & 15;
    const int klo  = half * 2;
    const int wave_m = wid >> 1;     // 0..3
    const int wave_n = wid & 1;      // 0..1

    const int rowBase = blockIdx.y * 128;
    const int colBase = blockIdx.x * 64;

    v8f acc[2][2] = {};

    const int lr = tid >> 2;         // 0..63
    const int lc = (tid & 3) << 2;   // 0,4,8,12

    for (int kb = 0; kb < Kdim; kb += 16) {
        __syncthreads();
        *(float4*)&As[lr * 20 + lc] =
            *(const float4*)(A + (size_t)(rowBase + lr) * Kdim + kb + lc);
        *(float4*)&As[(lr + 64) * 20 + lc] =
            *(const float4*)(A + (size_t)(rowBase + lr + 64) * Kdim + kb + lc);
        *(float4*)&Wsh[lr * 20 + lc] =
            *(const float4*)(W + (size_t)(colBase + lr) * Kdim + kb + lc);
        __syncthreads();

#pragma unroll
        for (int ks = 0; ks < 16; ks += 4) {
            v2f a0 = *(const v2f*)&As[(wave_m * 32 + l15) * 20 + ks + klo];
            v2f a1 = *(const v2f*)&As[(wave_m * 32 + 16 + l15) * 20 + ks + klo];
            v2f w0 = *(const v2f*)&Wsh[(wave_n * 32 + l15) * 20 + ks + klo];
            v2f w1 = *(const v2f*)&Wsh[(wave_n * 32 + 16 + l15) * 20 + ks + klo];
            acc[0][0] = wmma4(a0, w0, acc[0][0]);
            acc[0][1] = wmma4(a0, w1, acc[0][1]);
            acc[1][0] = wmma4(a1, w0, acc[1][0]);
            acc[1][1] = wmma4(a1, w1, acc[1][1]);
        }
    }

#pragma unroll
    for (int sm = 0; sm < 2; ++sm) {
#pragma unroll
        for (int sn = 0; sn < 2; ++sn) {
            const int j  = colBase + wave_n * 32 + sn * 16 + l15;
            const int r0 = rowBase + wave_m * 32 + sm * 16 + half * 8;
            if (MODE == 0) {
                const float bv = bias ? bias[j] : 0.0f;
                float* dst = out + (size_t)r0 * Ncols + j;
#pragma unroll
                for (int g = 0; g < 8; ++g)
                    dst[(size_t)g * Ncols] = acc[sm][sn][g] + bv;
            } else {
                // j = t*1024 + h*64 + d ; row r = b*1024 + n
                const int t  = j >> 10;
                const int hh = (j >> 6) & 15;
                const int d  = j & 63;
                const int bb = r0 >> 10;
                const int n  = r0 & 1023;
                float* dst;
                if (t == 0)
                    dst = Qb + ((size_t)(bb * 16 + hh) * 1024 + n) * 64 + d;
                else if (t == 1)
                    dst = Kb + ((size_t)(bb * 16 + hh) * 1280 + 256 + n) * 64 + d;
                else
                    dst = Vb + ((size_t)(bb * 16 + hh) * 1280 + 256 + n) * 64 + d;
#pragma unroll
                for (int g = 0; g < 8; ++g)
                    dst[(size_t)g * 64] = acc[sm][sn][g];
            }
        }
    }
}

// ---------------------------------------------------------------------------
// Scatter pre-projected context K,V (B,256,2C) into front of K/V (B,H,1280,D).
// ---------------------------------------------------------------------------
__global__ __launch_bounds__(256) void ctx_scatter_kernel(
    const float* __restrict__ ctx, float* __restrict__ Kb, float* __restrict__ Vb)
{
    const int idx = blockIdx.x * 256 + threadIdx.x;  // float4 index, 524288 total
    const int f   = idx << 2;
    const int b   = f >> 19;          // / (256*2048)
    const int m   = (f >> 11) & 255;  // kv position within context
    const int q   = f & 2047;         // channel within 2C
    const int s   = q >> 10;          // 0 = K, 1 = V
    const int hh  = (q >> 6) & 15;
    const int d   = q & 63;
    const float4 v = *(const float4*)(ctx + (size_t)f);
    float* base = s ? Vb : Kb;
    *(float4*)(base + ((size_t)(b * 16 + hh) * 1280 + m) * 64 + d) = v;
}

// ---------------------------------------------------------------------------
// Flash-style attention. Block = 8 waves x 16 query rows = 128 rows.
// Per kv-chunk of 32: S = (Q*scale) K^T via WMMA, online softmax with
// cross-lane (16-lane half) reductions, P re-layout through padded LDS,
// O += P V via WMMA against transposed V tile. Output written directly in
// (B, N, H*D) layout for the projection GEMM.
// ---------------------------------------------------------------------------
__global__ __launch_bounds__(256) void attn_kernel(
    const float* __restrict__ Qb, const float* __restrict__ Kb,
    const float* __restrict__ Vb, float* __restrict__ AO)
{
    __shared__ float Ks[32 * 68];        // K chunk, padded row stride
    __shared__ float VTs[64 * 34];       // V chunk transposed (d-major), padded
    __shared__ float Ps[8 * 16 * 34];    // per-wave P tile for A-frag re-layout

    const int tid  = threadIdx.x;
    const int wid  = tid >> 5;
    const int lane = tid & 31;
    const int half = lane >> 4;
    const int l15  = lane & 15;
    const int klo  = half * 2;

    const int bh = blockIdx.x;
    const int b  = bh >> 4;
    const int h  = bh & 15;
    const float* Qp = Qb + (size_t)bh * 1024 * 64;
    const float* Kp = Kb + (size_t)bh * 1280 * 64;
    const float* Vp = Vb + (size_t)bh * 1280 * 64;

    const int n0   = blockIdx.y * 128 + wid * 16;  // wave's first query row
    const int qrow = n0 + l15;
    const float scale = 0.125f;                    // D^-0.5, exact

    // Q rows as 16 A-fragments (full D=64), scale folded in.
    v2f qf[16];
#pragma unroll
    for (int kk = 0; kk < 16; ++kk) {
        v2f t = *(const v2f*)(Qp + (size_t)qrow * 64 + kk * 4 + klo);
        qf[kk] = t * scale;
    }

    float mrow[8], lrow[8];
    v8f oacc[4] = {};
#pragma unroll
    for (int g = 0; g < 8; ++g) { mrow[g] = -3.0e38f; lrow[g] = 0.0f; }

    float* Pw = &Ps[wid * 16 * 34];

    for (int mc = 0; mc < 1280; mc += 32) {
        __syncthreads();
        // Cooperative load: K chunk row-major, V chunk transposed.
#pragma unroll
        for (int it = 0; it < 2; ++it) {
            const int idx = tid + it * 256;   // 0..511
            const int r   = idx >> 4;         // kv row 0..31
            const int c4  = (idx & 15) << 2;  // d 0..60
            *(float4*)&Ks[r * 68 + c4] =
                *(const float4*)(Kp + (size_t)(mc + r) * 64 + c4);
            const float4 v = *(const float4*)(Vp + (size_t)(mc + r) * 64 + c4);
            VTs[(c4 + 0) * 34 + r] = v.x;
            VTs[(c4 + 1) * 34 + r] = v.y;
            VTs[(c4 + 2) * 34 + r] = v.z;
            VTs[(c4 + 3) * 34 + r] = v.w;
        }
        __syncthreads();

        // S chunk (16 rows x 32 kv-cols) = Q K^T
        v8f s0 = {}, s1 = {};
#pragma unroll
        for (int kk = 0; kk < 16; ++kk) {
            v2f b0 = *(const v2f*)&Ks[l15 * 68 + kk * 4 + klo];
            v2f b1 = *(const v2f*)&Ks[(16 + l15) * 68 + kk * 4 + klo];
            s0 = wmma4(qf[kk], b0, s0);
            s1 = wmma4(qf[kk], b1, s1);
        }

        // Online softmax: row stats reduced across the 16-lane half.
        float alpha[8];
#pragma unroll
        for (int g = 0; g < 8; ++g) {
            float cm = fmaxf(s0[g], s1[g]);
            cm = fmaxf(cm, __shfl_xor(cm, 1, 32));
            cm = fmaxf(cm, __shfl_xor(cm, 2, 32));
            cm = fmaxf(cm, __shfl_xor(cm, 4, 32));
            cm = fmaxf(cm, __shfl_xor(cm, 8, 32));
            const float mnew = fmaxf(mrow[g], cm);
            alpha[g] = __expf(mrow[g] - mnew);
            mrow[g]  = mnew;
            const float p0 = __expf(s0[g] - mnew);
            const float p1 = __expf(s1[g] - mnew);
            s0[g] = p0; s1[g] = p1;
            float cs = p0 + p1;
            cs += __shfl_xor(cs, 1, 32);
            cs += __shfl_xor(cs, 2, 32);
            cs += __shfl_xor(cs, 4, 32);
            cs += __shfl_xor(cs, 8, 32);
            lrow[g] = lrow[g] * alpha[g] + cs;
        }
#pragma unroll
        for (int t4 = 0; t4 < 4; ++t4)
#pragma unroll
            for (int g = 0; g < 8; ++g) oacc[t4][g] *= alpha[g];

        // Re-layout P (C-layout -> A-fragment layout) via per-wave LDS tile.
#pragma unroll
        for (int g = 0; g < 8; ++g) {
            const int r = g + half * 8;
            Pw[r * 34 + l15]      = s0[g];
            Pw[r * 34 + 16 + l15] = s1[g];
        }
        asm volatile("s_wait_dscnt 0" ::: "memory");

        // O += P V   (B-frags from transposed V tile)
#pragma unroll
        for (int kk = 0; kk < 8; ++kk) {
            v2f pf = *(const v2f*)&Pw[l15 * 34 + kk * 4 + klo];
#pragma unroll
            for (int t4 = 0; t4 < 4; ++t4) {
                v2f vf = *(const v2f*)&VTs[(t4 * 16 + l15) * 34 + kk * 4 + klo];
                oacc[t4] = wmma4(pf, vf, oacc[t4]);
            }
        }
    }

    // Normalize and write in (B, N, H*D) layout.
#pragma unroll
    for (int g = 0; g < 8; ++g) {
        const float rinv = 1.0f / lrow[g];
        const int n = n0 + g + half * 8;
        float* dst = AO + ((size_t)(b * 1024 + n) * 1024) + h * 64;
#pragma unroll
        for (int t4 = 0; t4 < 4; ++t4)
            dst[t4 * 16 + l15] = oacc[t4][g] * rinv;
    }
}

// ---------------------------------------------------------------------------
extern "C" void kernel_launch(void* const* d_in, const int* in_sizes, int n_in,
                              void* d_out, int out_size, void* d_ws, size_t ws_size,
                              hipStream_t stream)
{
    const float* x      = (const float*)d_in[0];
    const float* ctx    = (const float*)d_in[1];
    const float* w_qkv  = (const float*)d_in[2];
    const float* w_proj = (const float*)d_in[3];
    const float* b_proj = (const float*)d_in[4];
    float* outp = (float*)d_out;

    // Workspace: Q(16MB) K(20MB) V(20MB) AO(16MB)
    float* Qb = (float*)d_ws;
    float* Kb = Qb + 4194304;
    float* Vb = Kb + 5242880;
    float* AO = Vb + 5242880;

    // 1) QKV projection, scattered into attention layout (M=4096,N=3072,K=1024)
    gemm_kernel<1><<<dim3(3072 / 64, 4096 / 128), 256, 0, stream>>>(
        x, w_qkv, nullptr, nullptr, Qb, Kb, Vb, 3072, 1024);

    // 2) Context K,V into front of kv-sequence
    ctx_scatter_kernel<<<2048, 256, 0, stream>>>(ctx, Kb, Vb);

    // 3) Attention (B*H = 64 blocks x 8 row-tiles)
    attn_kernel<<<dim3(64, 8), 256, 0, stream>>>(Qb, Kb, Vb, AO);

    // 4) Output projection + bias (M=4096,N=1024,K=1024)
    gemm_kernel<0><<<dim3(1024 / 64, 4096 / 128), 256, 0, stream>>>(
        AO, w_proj, b_proj, outp, nullptr, nullptr, nullptr, 1024, 1024);
}